// GATGraphClassifier_30777735643797
// MI455X (gfx1250) — compile-verified
//
#include <hip/hip_runtime.h>
#include <math.h>

// ---------------------------------------------------------------------------
// GAT graph classifier for MI455X (gfx1250, wave32).
// Dense GEMMs run on V_WMMA_F32_16X16X4_F32 (full f32, matches reference).
// Sparse attention/aggregation uses flat edge-parallel kernels with
// GLOBAL_ATOMIC_ADD_F32 / MAX scatters that resolve in the 192 MB L2
// (h1=51MB, h2=26MB both L2-resident).
// ---------------------------------------------------------------------------

typedef __attribute__((ext_vector_type(2))) float v2f;
typedef __attribute__((ext_vector_type(8))) float v8f;

constexpr int NNODES = 50000;
constexpr int EDGES  = 800000;
constexpr int EP     = EDGES + NNODES;   // +self loops
constexpr int NGRAPH = 64;

// ---------------- utility ----------------
__global__ void k_fill(float* __restrict__ p, float v, unsigned n) {
  unsigned t = blockIdx.x * blockDim.x + threadIdx.x;
  if (t < n) p[t] = v;
}

__device__ __forceinline__ void atomicMaxF(float* addr, float val) {
  if (!(val < 0.0f)) {
    atomicMax(reinterpret_cast<int*>(addr), __float_as_int(val));
  } else {
    atomicMin(reinterpret_cast<unsigned int*>(addr), __float_as_uint(val));
  }
}

// ---------------- WMMA f32 GEMM: C[M,Nout] = A[M,K] * B[K,Nout] ------------
// One wave (32 lanes) computes one 16x16 tile; K consumed 4 per WMMA issue.
// A 16x4 layout : lanes 0-15 M=0..15 K={0,1} in v0/v1; lanes 16-31 K={2,3}.
// B 4x16 layout : v0 lanes0-15 K=0 row, lanes16-31 K=2 row; v1 -> K=1/K=3.
// C 16x16 layout: vgpr i, lanes0-15 M=i, lanes16-31 M=i+8.
__global__ void k_gemm_wmma(const float* __restrict__ A,
                            const float* __restrict__ B,
                            float* __restrict__ C,
                            int K, int Nout, int nTiles) {
  int tile  = blockIdx.x;
  int mBase = (tile / nTiles) << 4;
  int nBase = (tile % nTiles) << 4;
  int lane  = threadIdx.x & 31;
  int idx   = lane & 15;      // M index for A, N index for B/C
  int khalf = lane >> 4;      // 0 or 1

  const float* arow = A + (size_t)(mBase + idx) * K + 2 * khalf;
  const float* bcol = B + nBase + idx;

  v8f acc = {};
  for (int k0 = 0; k0 < K; k0 += 4) {
    v2f a, b;
    a.x = arow[k0];
    a.y = arow[k0 + 1];
    b.x = bcol[(size_t)(k0 + 2 * khalf) * Nout];
    b.y = bcol[(size_t)(k0 + 2 * khalf + 1) * Nout];
    acc = __builtin_amdgcn_wmma_f32_16x16x4_f32(
        /*neg_a=*/false, a, /*neg_b=*/false, b,
        /*c_mod=*/(short)0, acc, /*reuse_a=*/false, /*reuse_b=*/false);
  }
  float* crow = C + (size_t)(mBase + khalf * 8) * Nout + nBase + idx;
#pragma unroll
  for (int i = 0; i < 8; ++i) crow[(size_t)i * Nout] = acc[i];
}

// ---------------- per-node attention projections es/ed ---------------------
__global__ void k_att_scores(const float* __restrict__ hbuf,
                             const float* __restrict__ a_s,
                             const float* __restrict__ a_d,
                             float* __restrict__ es, float* __restrict__ ed,
                             int H, int C, unsigned total) {
  unsigned t = blockIdx.x * blockDim.x + threadIdx.x;
  if (t >= total) return;
  unsigned n = t / H;
  int h = t - n * H;
  const float* hp = hbuf + (size_t)n * H * C + (size_t)h * C;
  const float* ap = a_s + h * C;
  const float* dp = a_d + h * C;
  float s = 0.0f, d = 0.0f;
  for (int c = 0; c < C; ++c) {
    float v = hp[c];
    s += v * ap[c];
    d += v * dp[c];
  }
  es[t] = s;
  ed[t] = d;
}

// ---------------- edge pass 1: e = leaky_relu(es[src]+ed[dst]); seg-max ----
__global__ void k_edge_logit_max(const int* __restrict__ ei,
                                 const float* __restrict__ es,
                                 const float* __restrict__ ed,
                                 float* __restrict__ eatt,
                                 float* __restrict__ m,
                                 int H, unsigned total) {
  unsigned t = blockIdx.x * blockDim.x + threadIdx.x;
  if (t >= total) return;
  unsigned e = t / H;
  int h = t - e * H;
  int s, d;
  if (e < (unsigned)EDGES) { s = ei[e]; d = ei[EDGES + e]; }
  else                     { s = d = (int)(e - EDGES); }
  float v = es[(size_t)s * H + h] + ed[(size_t)d * H + h];
  v = (v > 0.0f) ? v : 0.2f * v;      // leaky_relu, slope 0.2
  eatt[t] = v;
  atomicMaxF(&m[(size_t)d * H + h], v);
}

// ---------------- edge pass 2: ex = exp(e - m[dst]); seg-sum ---------------
__global__ void k_edge_exp_sum(const int* __restrict__ ei,
                               float* __restrict__ eatt,
                               const float* __restrict__ m,
                               float* __restrict__ den,
                               int H, unsigned total) {
  unsigned t = blockIdx.x * blockDim.x + threadIdx.x;
  if (t >= total) return;
  unsigned e = t / H;
  int h = t - e * H;
  int d = (e < (unsigned)EDGES) ? ei[EDGES + e] : (int)(e - EDGES);
  float ex = expf(eatt[t] - m[(size_t)d * H + h]);
  eatt[t] = ex;
  atomicAdd(&den[(size_t)d * H + h], ex);
}

// ---------------- edge pass 3: alpha = ex / den[dst] -----------------------
__global__ void k_edge_norm(const int* __restrict__ ei,
                            float* __restrict__ eatt,
                            const float* __restrict__ den,
                            int H, unsigned total) {
  unsigned t = blockIdx.x * blockDim.x + threadIdx.x;
  if (t >= total) return;
  unsigned e = t / H;
  int h = t - e * H;
  int d = (e < (unsigned)EDGES) ? ei[EDGES + e] : (int)(e - EDGES);
  eatt[t] = eatt[t] / den[(size_t)d * H + h];
}

// ------------- aggregation: out[dst,c] += h[src,c] * alpha[e,head(c)] ------
__global__ void k_edge_aggregate(const int* __restrict__ ei,
                                 const float* __restrict__ hbuf,
                                 const float* __restrict__ alpha,
                                 float* __restrict__ out,
                                 int H, int hcShift, int headShift,
                                 unsigned total) {
  unsigned t = blockIdx.x * blockDim.x + threadIdx.x;
  if (t >= total) return;
  unsigned e = t >> hcShift;
  int c = t & ((1 << hcShift) - 1);
  int h = c >> headShift;
  int s, d;
  if (e < (unsigned)EDGES) { s = ei[e]; d = ei[EDGES + e]; }
  else                     { s = d = (int)(e - EDGES); }
  int HC = 1 << hcShift;
  atomicAdd(&out[(size_t)d * HC + c],
            hbuf[(size_t)s * HC + c] * alpha[(size_t)e * H + h]);
}

// ---------------- bias + ELU (in place) ------------------------------------
__global__ void k_bias_elu(float* __restrict__ buf, const float* __restrict__ bias,
                           int HC, unsigned total) {
  unsigned t = blockIdx.x * blockDim.x + threadIdx.x;
  if (t >= total) return;
  float v = buf[t] + bias[t % HC];
  buf[t] = (v > 0.0f) ? v : (expf(v) - 1.0f);
}

// ---------------- global mean pool -----------------------------------------
__global__ void k_pool_sum(const int* __restrict__ batch,
                           const float* __restrict__ feat,
                           float* __restrict__ pooled, unsigned total) {
  unsigned t = blockIdx.x * blockDim.x + threadIdx.x;
  if (t >= total) return;
  unsigned n = t >> 7;
  int c = t & 127;
  atomicAdd(&pooled[(size_t)batch[n] * 128 + c], feat[t]);
}

__global__ void k_pool_cnt(const int* __restrict__ batch,
                           float* __restrict__ cnt, unsigned n) {
  unsigned t = blockIdx.x * blockDim.x + threadIdx.x;
  if (t < n) atomicAdd(&cnt[batch[t]], 1.0f);
}

// ---------------- classifier head + log_softmax ----------------------------
__global__ void k_head(const float* __restrict__ pooled,
                       const float* __restrict__ cnt,
                       const float* __restrict__ fcw,
                       const float* __restrict__ fcb,
                       float* __restrict__ out) {
  int g = threadIdx.x;
  if (g >= NGRAPH) return;
  float inv = 1.0f / fmaxf(cnt[g], 1.0f);
  float logits[10];
  float mx = -3.4e38f;
  for (int j = 0; j < 10; ++j) {
    float acc = 0.0f;
    for (int c = 0; c < 128; ++c) acc += pooled[g * 128 + c] * fcw[c * 10 + j];
    float v = acc * inv + fcb[j];
    logits[j] = v;
    mx = fmaxf(mx, v);
  }
  float s = 0.0f;
  for (int j = 0; j < 10; ++j) s += expf(logits[j] - mx);
  float lse = mx + logf(s);
  for (int j = 0; j < 10; ++j) out[g * 10 + j] = logits[j] - lse;
}

// ---------------------------------------------------------------------------
static inline dim3 grid1(unsigned n) { return dim3((n + 255u) / 256u); }

extern "C" void kernel_launch(void* const* d_in, const int* in_sizes, int n_in,
                              void* d_out, int out_size, void* d_ws, size_t ws_size,
                              hipStream_t stream) {
  const float* x    = (const float*)d_in[0];
  const int*   ei   = (const int*)d_in[1];   // [2,E] flat: row0=src, row1=dst
  const int*   bat  = (const int*)d_in[2];
  const float* W1   = (const float*)d_in[3];
  const float* as1  = (const float*)d_in[4];
  const float* ad1  = (const float*)d_in[5];
  const float* b1   = (const float*)d_in[6];
  const float* W2   = (const float*)d_in[7];
  const float* as2  = (const float*)d_in[8];
  const float* ad2  = (const float*)d_in[9];
  const float* b2   = (const float*)d_in[10];
  const float* fcw  = (const float*)d_in[11];
  const float* fcb  = (const float*)d_in[12];
  float* out = (float*)d_out;

  // ---- carve workspace (floats), ~175 MB total ----
  float* w = (float*)d_ws;
  float* h1   = w; w += (size_t)NNODES * 256;   // x@W1
  float* act1 = w; w += (size_t)NNODES * 256;   // layer-1 aggregated + ELU
  float* h2   = w; w += (size_t)NNODES * 128;   // act1@W2
  float* act2 = w; w += (size_t)NNODES * 128;   // layer-2 aggregated + ELU
  float* e1   = w; w += (size_t)EP * 4;         // per-edge logits/alpha L1
  float* e2   = w; w += (size_t)EP;             // per-edge logits/alpha L2
  float* es1  = w; w += (size_t)NNODES * 4;
  float* ed1  = w; w += (size_t)NNODES * 4;
  float* m1   = w; w += (size_t)NNODES * 4;
  float* dn1  = w; w += (size_t)NNODES * 4;
  float* es2  = w; w += (size_t)NNODES;
  float* ed2  = w; w += (size_t)NNODES;
  float* m2   = w; w += (size_t)NNODES;
  float* dn2  = w; w += (size_t)NNODES;
  float* pooled = w; w += (size_t)NGRAPH * 128;
  float* cnt    = w; w += (size_t)NGRAPH;

  const float NEG_INF = -__builtin_huge_valf();

  // =================== Layer 1 (128 -> 4x64, concat) =======================
  k_gemm_wmma<<<(NNODES / 16) * 16, 32, 0, stream>>>(x, W1, h1, 128, 256, 16);
  k_att_scores<<<grid1(NNODES * 4), 256, 0, stream>>>(h1, as1, ad1, es1, ed1,
                                                      4, 64, NNODES * 4);
  k_fill<<<grid1(NNODES * 4), 256, 0, stream>>>(m1, NEG_INF, NNODES * 4);
  k_fill<<<grid1(NNODES * 4), 256, 0, stream>>>(dn1, 0.0f, NNODES * 4);
  k_fill<<<grid1(NNODES * 256), 256, 0, stream>>>(act1, 0.0f, NNODES * 256);

  k_edge_logit_max<<<grid1(EP * 4), 256, 0, stream>>>(ei, es1, ed1, e1, m1, 4, EP * 4);
  k_edge_exp_sum<<<grid1(EP * 4), 256, 0, stream>>>(ei, e1, m1, dn1, 4, EP * 4);
  k_edge_norm<<<grid1(EP * 4), 256, 0, stream>>>(ei, e1, dn1, 4, EP * 4);
  k_edge_aggregate<<<grid1((unsigned)EP * 256u), 256, 0, stream>>>(
      ei, h1, e1, act1, 4, /*hcShift=*/8, /*headShift=*/6, (unsigned)EP * 256u);
  k_bias_elu<<<grid1(NNODES * 256), 256, 0, stream>>>(act1, b1, 256, NNODES * 256);

  // =================== Layer 2 (256 -> 1x128, mean==identity) ==============
  k_gemm_wmma<<<(NNODES / 16) * 8, 32, 0, stream>>>(act1, W2, h2, 256, 128, 8);
  k_att_scores<<<grid1(NNODES), 256, 0, stream>>>(h2, as2, ad2, es2, ed2,
                                                  1, 128, NNODES);
  k_fill<<<grid1(NNODES), 256, 0, stream>>>(m2, NEG_INF, NNODES);
  k_fill<<<grid1(NNODES), 256, 0, stream>>>(dn2, 0.0f, NNODES);
  k_fill<<<grid1(NNODES * 128), 256, 0, stream>>>(act2, 0.0f, NNODES * 128);

  k_edge_logit_max<<<grid1(EP), 256, 0, stream>>>(ei, es2, ed2, e2, m2, 1, EP);
  k_edge_exp_sum<<<grid1(EP), 256, 0, stream>>>(ei, e2, m2, dn2, 1, EP);
  k_edge_norm<<<grid1(EP), 256, 0, stream>>>(ei, e2, dn2, 1, EP);
  k_edge_aggregate<<<grid1((unsigned)EP * 128u), 256, 0, stream>>>(
      ei, h2, e2, act2, 1, /*hcShift=*/7, /*headShift=*/7, (unsigned)EP * 128u);
  k_bias_elu<<<grid1(NNODES * 128), 256, 0, stream>>>(act2, b2, 128, NNODES * 128);

  // =================== Pool + head ========================================
  k_fill<<<grid1(NGRAPH * 128 + NGRAPH), 256, 0, stream>>>(pooled, 0.0f,
                                                           NGRAPH * 128 + NGRAPH);
  k_pool_sum<<<grid1(NNODES * 128), 256, 0, stream>>>(bat, act2, pooled,
                                                      NNODES * 128);
  k_pool_cnt<<<grid1(NNODES), 256, 0, stream>>>(bat, cnt, NNODES);
  k_head<<<1, 64, 0, stream>>>(pooled, cnt, fcw, fcb, out);
}